// BaseBidirectionalAttention_69612829934279
// MI455X (gfx1250) — compile-verified
//
#include <hip/hip_runtime.h>

// BiDAF bidirectional attention for MI455X (gfx1250, wave32).
// GEMMs on V_WMMA_F32_16X16X4_F32; global->LDS staging via the CDNA5 async
// copy engine (GLOBAL_LOAD_ASYNC_TO_LDS_B128, ASYNCcnt) where available.

typedef float v2f __attribute__((ext_vector_type(2)));
typedef float v8f __attribute__((ext_vector_type(8)));
typedef int   v4i __attribute__((__vector_size__(16)));   // int4 element type for async builtin

#define B_  32
#define C_  1024
#define Q_  64
#define E_  256
#define FE_ 1024
#define AS  68    // LDS row stride (floats) for [row][k] tiles: bank = 4*row+k, conflict-free
#define QS  136   // LDS row stride for [k][n] tiles: 2*QS % 64 == 16 -> halves on disjoint banks

#if __has_builtin(__builtin_amdgcn_global_load_async_to_lds_b128)
#define HAS_ASYNC_LDS 1
#else
#define HAS_ASYNC_LDS 0
#endif

static __device__ __forceinline__ void copy16_g2l(float* lds, const float* g) {
#if HAS_ASYNC_LDS
  __builtin_amdgcn_global_load_async_to_lds_b128(
      (__attribute__((address_space(1))) v4i*)(void*)g,
      (__attribute__((address_space(3))) v4i*)(void*)lds, 0, 0);
#else
  *(float4*)lds = *(const float4*)g;
#endif
}

static __device__ __forceinline__ void async_wait() {
#if HAS_ASYNC_LDS
#if __has_builtin(__builtin_amdgcn_s_wait_asynccnt)
  __builtin_amdgcn_s_wait_asynccnt(0);
#else
  asm volatile("s_wait_asynccnt 0x0" ::: "memory");
#endif
#endif
}

static __device__ __forceinline__ v8f wmma4(v2f a, v2f b, v8f c) {
  // D = A(16x4 f32) * B(4x16 f32) + C(16x16 f32)
  return __builtin_amdgcn_wmma_f32_16x16x4_f32(false, a, false, b, (short)0, c, false, false);
}
static __device__ __forceinline__ v8f vzero() {
  v8f z = {0.f, 0.f, 0.f, 0.f, 0.f, 0.f, 0.f, 0.f};
  return z;
}

// ---------------------------------------------------------------------------
// K0: ctx_weighted[b,c] = <context[b,c,:], w_context>; q_weighted likewise.
// ---------------------------------------------------------------------------
__launch_bounds__(256)
__global__ void k0_rowdots(const float* __restrict__ ctx, const float* __restrict__ qst,
                           const float* __restrict__ w_c, const float* __restrict__ w_q,
                           float* __restrict__ ctx_w, float* __restrict__ q_w) {
  int gid = blockIdx.x * 256 + threadIdx.x;
  const int nCtx = B_ * C_;
  if (gid < nCtx) {
    const float* row = ctx + (size_t)gid * E_;
    float acc = 0.f;
#pragma unroll 4
    for (int e = 0; e < E_; e += 4) {
      float4 r = *(const float4*)(row + e);
      float4 w = *(const float4*)(w_c + e);
      acc += r.x * w.x + r.y * w.y + r.z * w.z + r.w * w.w;
    }
    ctx_w[gid] = acc;
  } else {
    int qid = gid - nCtx;
    if (qid < B_ * Q_) {
      const float* row = qst + (size_t)qid * E_;
      float acc = 0.f;
#pragma unroll 4
      for (int e = 0; e < E_; e += 4) {
        float4 r = *(const float4*)(row + e);
        float4 w = *(const float4*)(w_q + e);
        acc += r.x * w.x + r.y * w.y + r.z * w.z + r.w * w.w;
      }
      q_w[qid] = acc;
    }
  }
}

// ---------------------------------------------------------------------------
// K1: similarity tile (64 c-rows x 64 q) via WMMA, + scalar terms, softmax over q.
// grid = (C/64, B), block = 256 (8 waves; wave w -> m-tile w%4, n-tiles (w/4)*2+{0,1})
// ---------------------------------------------------------------------------
__launch_bounds__(256)
__global__ void k1_sim_softmax(const float* __restrict__ ctx, const float* __restrict__ qst,
                               const float* __restrict__ w_m,
                               const float* __restrict__ ctx_w, const float* __restrict__ q_w,
                               float* __restrict__ P, float* __restrict__ rowmax) {
  __shared__ float Alds[64 * AS];   // (ctx * w_mult)[m][k]
  __shared__ float Blds[64 * AS];   // question[q][k] == B stored as [n][k]
  __shared__ float Slds[64 * AS];   // similarity tile [c][q]
  __shared__ float qwbuf[64];

  const int ct = blockIdx.x, b = blockIdx.y;
  const int c0 = ct * 64;
  const int tid = threadIdx.x;
  const int wave = tid >> 5, lane = tid & 31, half = lane >> 4, lr = lane & 15;
  const int wm = wave & 3;
  const int wn0 = (wave >> 2) * 2;

  v8f acc0 = vzero(), acc1 = vzero();

  for (int chunk = 0; chunk < 4; ++chunk) {
    const int e0 = chunk * 64;
    __syncthreads();
    // B = question chunk (pure copy): async DMA into LDS
#pragma unroll
    for (int i = 0; i < 4; ++i) {
      int j = tid + i * 256, r = j >> 4, c4 = (j & 15) * 4;
      copy16_g2l(&Blds[r * AS + c4], qst + ((size_t)b * Q_ + r) * E_ + e0 + c4);
    }
    // A = ctx * w_mult: two-phase (batch loads, then stores)
    float4 cv[4], wv[4];
#pragma unroll
    for (int i = 0; i < 4; ++i) {
      int j = tid + i * 256, r = j >> 4, c4 = (j & 15) * 4;
      cv[i] = *(const float4*)(ctx + ((size_t)b * C_ + c0 + r) * E_ + e0 + c4);
      wv[i] = *(const float4*)(w_m + e0 + c4);
    }
#pragma unroll
    for (int i = 0; i < 4; ++i) {
      int j = tid + i * 256, r = j >> 4, c4 = (j & 15) * 4;
      float4 av;
      av.x = cv[i].x * wv[i].x; av.y = cv[i].y * wv[i].y;
      av.z = cv[i].z * wv[i].z; av.w = cv[i].w * wv[i].w;
      *(float4*)(&Alds[r * AS + c4]) = av;
    }
    async_wait();
    __syncthreads();
#pragma unroll
    for (int kk = 0; kk < 64; kk += 4) {
      v2f a;
      a.x = Alds[(wm * 16 + lr) * AS + kk + 2 * half];
      a.y = Alds[(wm * 16 + lr) * AS + kk + 2 * half + 1];
      v2f b0, b1;
      b0.x = Blds[(wn0 * 16 + lr) * AS + kk + 2 * half];
      b0.y = Blds[(wn0 * 16 + lr) * AS + kk + 2 * half + 1];
      b1.x = Blds[((wn0 + 1) * 16 + lr) * AS + kk + 2 * half];
      b1.y = Blds[((wn0 + 1) * 16 + lr) * AS + kk + 2 * half + 1];
      acc0 = wmma4(a, b0, acc0);
      acc1 = wmma4(a, b1, acc1);
    }
  }
  // store D tiles to Slds (C/D layout: VGPR r -> row r + 8*half)
#pragma unroll
  for (int r = 0; r < 8; ++r) {
    Slds[(wm * 16 + r + 8 * half) * AS + wn0 * 16 + lr] = acc0[r];
    Slds[(wm * 16 + r + 8 * half) * AS + (wn0 + 1) * 16 + lr] = acc1[r];
  }
  if (tid < 64) qwbuf[tid] = q_w[b * Q_ + tid];
  __syncthreads();

  if (tid < 64) {
    const float cw = ctx_w[b * C_ + c0 + tid];
    float m = -1e30f;
#pragma unroll 8
    for (int q = 0; q < 64; ++q) {
      float s = Slds[tid * AS + q] + cw + qwbuf[q];
      Slds[tid * AS + q] = s;
      m = fmaxf(m, s);
    }
    rowmax[b * C_ + c0 + tid] = m;
    float sum = 0.f;
#pragma unroll 8
    for (int q = 0; q < 64; ++q) {
      float e = __expf(Slds[tid * AS + q] - m);
      Slds[tid * AS + q] = e;
      sum += e;
    }
    const float inv = 1.0f / sum;
    float* Prow = P + ((size_t)b * C_ + c0 + tid) * Q_;
#pragma unroll 8
    for (int q = 0; q < 64; ++q) Prow[q] = Slds[tid * AS + q] * inv;
  }
}

// ---------------------------------------------------------------------------
// K2: q2c_weights = softmax_c(rowmax[b,:]); q2c_att[b,e] = sum_c w[c]*ctx[b,c,e]
// ---------------------------------------------------------------------------
__launch_bounds__(256)
__global__ void k2_q2c(const float* __restrict__ ctx, const float* __restrict__ rowmax,
                       float* __restrict__ q2c) {
  __shared__ float w[C_];
  __shared__ float red[256];
  const int b = blockIdx.x, tid = threadIdx.x;

  float m = -1e30f;
#pragma unroll
  for (int i = 0; i < 4; ++i) {
    float v = rowmax[b * C_ + tid + i * 256];
    w[tid + i * 256] = v;
    m = fmaxf(m, v);
  }
  red[tid] = m;
  __syncthreads();
  for (int s = 128; s > 0; s >>= 1) {
    if (tid < s) red[tid] = fmaxf(red[tid], red[tid + s]);
    __syncthreads();
  }
  m = red[0];
  __syncthreads();

  float sum = 0.f;
#pragma unroll
  for (int i = 0; i < 4; ++i) {
    float e = __expf(w[tid + i * 256] - m);
    w[tid + i * 256] = e;
    sum += e;
  }
  red[tid] = sum;
  __syncthreads();
  for (int s = 128; s > 0; s >>= 1) {
    if (tid < s) red[tid] += red[tid + s];
    __syncthreads();
  }
  const float inv = 1.0f / red[0];

  float acc = 0.f;
  const float* cb = ctx + (size_t)b * C_ * E_ + tid;
#pragma unroll 4
  for (int c = 0; c < C_; ++c) acc += w[c] * cb[(size_t)c * E_];
  q2c[b * E_ + tid] = acc * inv;
}

// ---------------------------------------------------------------------------
// K3: c2q_att = P @ question (per batch M=1024, N=256, K=64) via WMMA.
// grid = (E/128, C/64, B); block = 256.
// ---------------------------------------------------------------------------
__launch_bounds__(256)
__global__ void k3_c2q(const float* __restrict__ P, const float* __restrict__ qst,
                       float* __restrict__ c2q) {
  __shared__ float Plds[64 * AS];   // [m=c][k=q]
  __shared__ float Qlds[64 * QS];   // [k=q][n=e]
  const int nt = blockIdx.x, ct = blockIdx.y, b = blockIdx.z;
  const int c0 = ct * 64, n0g = nt * 128;
  const int tid = threadIdx.x;
  const int wave = tid >> 5, lane = tid & 31, half = lane >> 4, lr = lane & 15;
  const int wm = wave & 3, wnb = (wave >> 2) * 4;

  // Pure copies: async DMA into LDS
#pragma unroll
  for (int i = 0; i < 4; ++i) {
    int j = tid + i * 256, r = j >> 4, c4 = (j & 15) * 4;
    copy16_g2l(&Plds[r * AS + c4], P + ((size_t)b * C_ + c0 + r) * Q_ + c4);
  }
#pragma unroll
  for (int i = 0; i < 8; ++i) {
    int j = tid + i * 256, r = j >> 5, c4 = (j & 31) * 4;
    copy16_g2l(&Qlds[r * QS + c4], qst + ((size_t)b * Q_ + r) * E_ + n0g + c4);
  }
  async_wait();
  __syncthreads();

  v8f acc[4];
#pragma unroll
  for (int t = 0; t < 4; ++t) acc[t] = vzero();

#pragma unroll
  for (int kk = 0; kk < 64; kk += 4) {
    v2f a;
    a.x = Plds[(wm * 16 + lr) * AS + kk + 2 * half];
    a.y = Plds[(wm * 16 + lr) * AS + kk + 2 * half + 1];
#pragma unroll
    for (int t = 0; t < 4; ++t) {
      int nn = (wnb + t) * 16;
      v2f bf;
      bf.x = Qlds[(kk + 2 * half) * QS + nn + lr];
      bf.y = Qlds[(kk + 2 * half + 1) * QS + nn + lr];
      acc[t] = wmma4(a, bf, acc[t]);
    }
  }
#pragma unroll
  for (int t = 0; t < 4; ++t) {
#pragma unroll
    for (int r = 0; r < 8; ++r) {
      int rowL = wm * 16 + r + 8 * half;
      int col = n0g + (wnb + t) * 16 + lr;
      c2q[((size_t)b * C_ + c0 + rowL) * E_ + col] = acc[t][r];
    }
  }
}

// ---------------------------------------------------------------------------
// K4: out = (attended @ final_W.T + b) * mask; attended built on the fly.
// Per batch: M=1024, N=1024, K=1024. grid = (FE/128, C/64, B), block = 256.
// ---------------------------------------------------------------------------
__launch_bounds__(256)
__global__ void k4_final(const float* __restrict__ ctx, const float* __restrict__ c2q,
                         const float* __restrict__ q2c, const float* __restrict__ W,
                         const float* __restrict__ bias, const float* __restrict__ mask,
                         float* __restrict__ out) {
  __shared__ float Alds[64 * AS];    // attended chunk [m][k]
  __shared__ float Wlds[128 * AS];   // final_W rows [n=f][k]
  __shared__ float q2cbuf[E_];
  __shared__ float bbuf[128];
  __shared__ float mbuf[64];

  const int nt = blockIdx.x, ct = blockIdx.y, b = blockIdx.z;
  const int f0 = nt * 128, c0 = ct * 64;
  const int tid = threadIdx.x;
  const int wave = tid >> 5, lane = tid & 31, half = lane >> 4, lr = lane & 15;
  const int wm = wave & 3, wnb = (wave >> 2) * 4;

  q2cbuf[tid] = q2c[b * E_ + tid];
  if (tid < 128) bbuf[tid] = bias[f0 + tid];
  if (tid < 64)  mbuf[tid] = mask[b * C_ + c0 + tid];

  v8f acc[4];
#pragma unroll
  for (int t = 0; t < 4; ++t) acc[t] = vzero();

  for (int chunk = 0; chunk < 16; ++chunk) {
    const int seg = chunk >> 2;          // attended segment 0..3
    const int e0 = (chunk & 3) * 64;     // offset within E
    const int k0 = chunk * 64;           // offset within FE (for W)
    __syncthreads();
    // stage W rows (pure copy): async DMA
#pragma unroll
    for (int i = 0; i < 8; ++i) {
      int j = tid + i * 256, r = j >> 4, c4 = (j & 15) * 4;
      copy16_g2l(&Wlds[r * AS + c4], W + (size_t)(f0 + r) * FE_ + k0 + c4);
    }
    // build attended chunk
    if (seg == 0) {
#pragma unroll
      for (int i = 0; i < 4; ++i) {
        int j = tid + i * 256, r = j >> 4, c4 = (j & 15) * 4;
        copy16_g2l(&Alds[r * AS + c4], ctx + ((size_t)b * C_ + c0 + r) * E_ + e0 + c4);
      }
    } else if (seg == 1) {
#pragma unroll
      for (int i = 0; i < 4; ++i) {
        int j = tid + i * 256, r = j >> 4, c4 = (j & 15) * 4;
        copy16_g2l(&Alds[r * AS + c4], c2q + ((size_t)b * C_ + c0 + r) * E_ + e0 + c4);
      }
    } else if (seg == 2) {
      float4 cv[4], qv[4];
#pragma unroll
      for (int i = 0; i < 4; ++i) {
        int j = tid + i * 256, r = j >> 4, c4 = (j & 15) * 4;
        size_t base = ((size_t)b * C_ + c0 + r) * E_ + e0 + c4;
        cv[i] = *(const float4*)(ctx + base);
        qv[i] = *(const float4*)(c2q + base);
      }
#pragma unroll
      for (int i = 0; i < 4; ++i) {
        int j = tid + i * 256, r = j >> 4, c4 = (j & 15) * 4;
        float4 av;
        av.x = cv[i].x * qv[i].x; av.y = cv[i].y * qv[i].y;
        av.z = cv[i].z * qv[i].z; av.w = cv[i].w * qv[i].w;
        *(float4*)(&Alds[r * AS + c4]) = av;
      }
    } else {
      float4 cv[4];
#pragma unroll
      for (int i = 0; i < 4; ++i) {
        int j = tid + i * 256, r = j >> 4, c4 = (j & 15) * 4;
        cv[i] = *(const float4*)(ctx + ((size_t)b * C_ + c0 + r) * E_ + e0 + c4);
      }
#pragma unroll
      for (int i = 0; i < 4; ++i) {
        int j = tid + i * 256, r = j >> 4, c4 = (j & 15) * 4;
        float4 av;
        av.x = cv[i].x * q2cbuf[e0 + c4 + 0];
        av.y = cv[i].y * q2cbuf[e0 + c4 + 1];
        av.z = cv[i].z * q2cbuf[e0 + c4 + 2];
        av.w = cv[i].w * q2cbuf[e0 + c4 + 3];
        *(float4*)(&Alds[r * AS + c4]) = av;
      }
    }
    async_wait();
    __syncthreads();
#pragma unroll
    for (int kk = 0; kk < 64; kk += 4) {
      v2f a;
      a.x = Alds[(wm * 16 + lr) * AS + kk + 2 * half];
      a.y = Alds[(wm * 16 + lr) * AS + kk + 2 * half + 1];
#pragma unroll
      for (int t = 0; t < 4; ++t) {
        int nn = (wnb + t) * 16;
        v2f bf;
        bf.x = Wlds[(nn + lr) * AS + kk + 2 * half];
        bf.y = Wlds[(nn + lr) * AS + kk + 2 * half + 1];
        acc[t] = wmma4(a, bf, acc[t]);
      }
    }
  }
  // epilogue: bias + mask, store
#pragma unroll
  for (int t = 0; t < 4; ++t) {
#pragma unroll
    for (int r = 0; r < 8; ++r) {
      int rowL = wm * 16 + r + 8 * half;
      int colL = (wnb + t) * 16 + lr;
      out[((size_t)b * C_ + c0 + rowL) * FE_ + f0 + colL] =
          (acc[t][r] + bbuf[colL]) * mbuf[rowL];
    }
  }
}

// ---------------------------------------------------------------------------
extern "C" void kernel_launch(void* const* d_in, const int* in_sizes, int n_in,
                              void* d_out, int out_size, void* d_ws, size_t ws_size,
                              hipStream_t stream) {
  (void)in_sizes; (void)n_in; (void)out_size; (void)ws_size;
  const float* ctx  = (const float*)d_in[0];  // (B,C,E)
  const float* qst  = (const float*)d_in[1];  // (B,Q,E)
  const float* mask = (const float*)d_in[2];  // (B,C)
  const float* w_q  = (const float*)d_in[3];  // (E,)
  const float* w_c  = (const float*)d_in[4];  // (E,)
  const float* w_m  = (const float*)d_in[5];  // (E,)
  const float* W    = (const float*)d_in[6];  // (FE,FE)
  const float* bias = (const float*)d_in[7];  // (FE,)
  float* out = (float*)d_out;                 // (B,C,FE)

  float* ws     = (float*)d_ws;
  float* c2q    = ws;                                   // B*C*E   = 8,388,608
  float* P      = c2q    + (size_t)B_ * C_ * E_;        // B*C*Q   = 2,097,152
  float* rowmax = P      + (size_t)B_ * C_ * Q_;        // B*C     = 32,768
  float* ctxw   = rowmax + (size_t)B_ * C_;             // B*C     = 32,768
  float* qw     = ctxw   + (size_t)B_ * C_;             // B*Q     = 2,048
  float* q2c    = qw     + (size_t)B_ * Q_;             // B*E     = 8,192

  k0_rowdots<<<(B_ * C_ + B_ * Q_ + 255) / 256, 256, 0, stream>>>(ctx, qst, w_c, w_q, ctxw, qw);
  k1_sim_softmax<<<dim3(C_ / 64, B_), 256, 0, stream>>>(ctx, qst, w_m, ctxw, qw, P, rowmax);
  k2_q2c<<<B_, 256, 0, stream>>>(ctx, rowmax, q2c);
  k3_c2q<<<dim3(E_ / 128, C_ / 64, B_), 256, 0, stream>>>(P, qst, c2q);
  k4_final<<<dim3(FE_ / 128, C_ / 64, B_), 256, 0, stream>>>(ctx, c2q, q2c, W, bias, mask, out);
}